// WeightedDistanceTransform_64527588655160
// MI455X (gfx1250) — compile-verified
//
#include <hip/hip_runtime.h>
#include <stdint.h>

// Weighted Euclidean distance transform (exact replication of the JAX ref):
//   K1: row min-scans (exact 1D dist) -> g^2, written transposed to scratch
//   K2: brute-force min-plus column pass d2[i,j] = min_k g2[k,j] + (i-k)^2,
//       sqrt, unnormalized d -> d_out, per-slice max via atomicMax(float bits)
//   K3: out = (d / max) * weight[c]   (reference rounding order)
//
// CDNA5/MI455X specifics: async global->LDS copies (ASYNCcnt) for all staging,
// 266KB LDS-resident slice in K1 (needs the 320KB WGP LDS), conflict-free
// stride-257 LDS layout in the VALU-bound K2 hot loop, wave32 blocks.

#define Hdim 256
#define Wdim 256
#define S1   260      // K1 LDS row stride (floats): rows 16B-aligned for b128 async
#define S2   257      // K2 LDS column stride (floats): conflict-free ds_load in hot loop
#define BIGF 1.0e6f
#define JT   32       // columns per K2 block
#define NSLICE_ELEMS (Hdim * Wdim)

// ws layout (floats): [0..11] per-slice max (float bits, uint); [64..] g2^T (12*65536)
#define WS_MAX_OFF 0
#define WS_G2T_OFF 64

static __device__ __forceinline__ uint32_t lds_addr_of(const void* p) {
    // generic pointer to LDS: low 32 bits == wave-relative LDS byte offset
    return (uint32_t)(uint64_t)(uintptr_t)p;
}

// --------------------------- K1: row pass -----------------------------------
__global__ __launch_bounds__(256) void wdt_rows(const float* __restrict__ src,
                                                float* __restrict__ ws)
{
    __shared__ float tile[Hdim * S1];          // 266,240 B: full slice resident
    const int tid   = threadIdx.x;
    const int slice = blockIdx.x;
    const size_t base = (size_t)slice * NSLICE_ELEMS;

    if (tid == 0) ((uint32_t*)ws)[WS_MAX_OFF + slice] = 0u;   // init slice max

    // ---- async global->LDS staging: 16B/lane/issue, 64 issues = 256KB ----
    uint32_t laddr = lds_addr_of(&tile[0]) +
                     (uint32_t)((((tid >> 6) * S1) + ((tid & 63) << 2)) * 4);
    uint64_t gaddr = (uint64_t)(uintptr_t)(src + base + (size_t)tid * 4);
    for (int it = 0; it < 64; ++it) {
        asm volatile("global_load_async_to_lds_b128 %0, %1, off"
                     :: "v"(laddr), "v"(gaddr) : "memory");
        laddr += 4u * S1 * 4u;     // advance 4 LDS rows
        gaddr += 4u * Wdim * 4u;   // advance 4 global rows
    }
    asm volatile("s_wait_asynccnt 0x0" ::: "memory");
    __syncthreads();

    // ---- exact 1D row distance: forward scan, backward scan, g2 in LDS ----
    float* rowp = &tile[tid * S1];
    float carry = BIGF;
    #pragma unroll 4
    for (int jj = 0; jj < Wdim; ++jj) {
        float f0 = (rowp[jj] == 0.0f) ? 0.0f : BIGF;
        float d  = fminf(f0, carry + 1.0f);
        rowp[jj] = d;
        carry = d;
    }
    carry = BIGF;
    #pragma unroll 4
    for (int jj = Wdim - 1; jj >= 0; --jj) {
        float d  = fminf(rowp[jj], carry + 1.0f);
        rowp[jj] = d * d;                       // g2
        carry = d;
    }

    // ---- write g2 transposed (coalesced: lanes are consecutive rows) ------
    float* g2T = ws + WS_G2T_OFF + base + tid;  // + jj*Hdim below
    #pragma unroll 4
    for (int jj = 0; jj < Wdim; ++jj)
        g2T[(size_t)jj * Hdim] = rowp[jj];
}

// --------------------------- K2: column pass --------------------------------
__global__ __launch_bounds__(256) void wdt_cols(float* __restrict__ ws,
                                                float* __restrict__ out)
{
    __shared__ float tile2[JT * S2];            // 32,896 B
    __shared__ float red[256];
    const int tid   = threadIdx.x;
    const int slice = blockIdx.x >> 3;          // Wdim/JT == 8 tiles per slice
    const int jt    = blockIdx.x & 7;
    const int jj    = tid & (JT - 1);           // column within tile
    const int iq    = tid >> 5;                 // 0..7: 32-row i-subrange
    const float* g2T = ws + WS_G2T_OFF + (size_t)slice * NSLICE_ELEMS
                          + (size_t)jt * JT * Hdim;

    // ---- async staging of 32 contiguous g2^T columns (b32: 4B alignment) --
    uint32_t laddr = lds_addr_of(&tile2[0]) + (uint32_t)(tid * 4);
    uint64_t gaddr = (uint64_t)(uintptr_t)(g2T + tid);
    for (int it = 0; it < JT; ++it) {
        asm volatile("global_load_async_to_lds_b32 %0, %1, off"
                     :: "v"(laddr), "v"(gaddr) : "memory");
        laddr += S2 * 4u;          // next LDS column (stride 257 floats)
        gaddr += Hdim * 4u;        // next global column (256 floats)
    }
    asm volatile("s_wait_asynccnt 0x0" ::: "memory");
    __syncthreads();

    // ---- min-plus: 16 register accumulators per k-sweep, conflict-free ----
    const int   jcol = jt * JT + jj;
    float*      outc = out + (size_t)slice * NSLICE_ELEMS + jcol;
    const float* colp = &tile2[jj * S2];
    float lmax = 0.0f;
    for (int half = 0; half < 2; ++half) {
        const int i0 = iq * 32 + half * 16;
        float acc[16];
        #pragma unroll
        for (int u = 0; u < 16; ++u) acc[u] = 3.0e38f;
        for (int k = 0; k < Hdim; ++k) {
            float g2 = colp[k];                 // lanes hit distinct banks
            float q0 = (float)(i0 - k);
            #pragma unroll
            for (int u = 0; u < 16; ++u) {
                float q = q0 + (float)u;        // integer-valued: fma == mul+add
                acc[u] = fminf(acc[u], __builtin_fmaf(q, q, g2));
            }
        }
        #pragma unroll
        for (int u = 0; u < 16; ++u) {
            float d = __builtin_sqrtf(acc[u]);
            lmax = fmaxf(lmax, d);
            outc[(size_t)(i0 + u) * Wdim] = d;  // coalesced per i-row
        }
    }

    // ---- block max reduce, one atomic per block (order-independent) -------
    red[tid] = lmax;
    __syncthreads();
    for (int s = 128; s > 0; s >>= 1) {
        if (tid < s) red[tid] = fmaxf(red[tid], red[tid + s]);
        __syncthreads();
    }
    if (tid == 0)
        atomicMax(((uint32_t*)ws) + WS_MAX_OFF + slice, __float_as_uint(red[0]));
}

// --------------------------- K3: normalize + weight -------------------------
__global__ __launch_bounds__(256) void wdt_norm(float* __restrict__ out,
                                                const float* __restrict__ wts,
                                                const float* __restrict__ ws,
                                                int channels)
{
    const int idx   = blockIdx.x * 256 + threadIdx.x;
    const int slice = idx >> 16;                // 65536 elems per slice
    float m  = __uint_as_float(((const uint32_t*)ws)[WS_MAX_OFF + slice]);
    float mm = (m > 0.0f) ? m : 1.0f;
    float wc = wts[slice % channels];           // uniform -> scalar load
    out[idx] = (out[idx] / mm) * wc;            // reference rounding order
}

// ---------------------------------------------------------------------------
extern "C" void kernel_launch(void* const* d_in, const int* in_sizes, int n_in,
                              void* d_out, int out_size, void* d_ws, size_t ws_size,
                              hipStream_t stream) {
    (void)n_in; (void)ws_size; (void)out_size;
    const float* flood = (const float*)d_in[0];
    const float* w     = (const float*)d_in[1];
    float* out         = (float*)d_out;
    float* ws          = (float*)d_ws;          // needs 256B + 3MB (<< ws_size)

    const int channels = in_sizes[1];                      // 3
    const int slices   = in_sizes[0] / NSLICE_ELEMS;       // B*C = 12

    wdt_rows<<<dim3(slices),               dim3(256), 0, stream>>>(flood, ws);
    wdt_cols<<<dim3(slices * (Wdim / JT)), dim3(256), 0, stream>>>(ws, out);
    wdt_norm<<<dim3(slices * NSLICE_ELEMS / 256), dim3(256), 0, stream>>>(
        out, w, ws, channels);
}